// PolicyNet_15247133900855
// MI455X (gfx1250) — compile-verified
//
#include <hip/hip_runtime.h>
#include <math.h>

// Problem constants (match reference)
#define NN   100000
#define EE   1600000
#define BB   4
#define PP   (NN / BB)     // 25000
#define FF   16
#define HH   128
#define AV_  64
#define HALF_ 32
#define AA   50
#define KK   4

typedef __attribute__((ext_vector_type(2))) float v2f;
typedef __attribute__((ext_vector_type(8))) float v8f;

// ---------------------------------------------------------------------------
// GEMM: Y[M x NOUT] = act(X)[M x KD] @ W[NOUT x KD]^T (+ bias) (+ relu-out)
//   act(X) = RELU_IN ? max(X,0) : X   (fuses the previous layer's ReLU)
// One wave computes a 16 x NOUT stripe using V_WMMA_F32_16X16X4_F32.
//
// A-frag (16x4 f32): lane = 16*khalf + m ; v[j] = X[m0+m][k0 + 2*khalf + j]
// B-frag (4x16 f32): lane = 16*khalf + n ; v[j] = W^T[k0+2*khalf+j][n0+n]
//                                               = W[(n0+n)*KD + k0+2*khalf+j]
// C/D (16x16 f32, 8 VGPRs): acc[r] @ lane(16*khalf+n) = C[m0 + r + 8*khalf][n0+n]
// ---------------------------------------------------------------------------
template <int KD, int NT, bool BIAS, bool RELU_IN, bool RELU_OUT>
__global__ __launch_bounds__(256) void gemm_wmma_kernel(
    const float* __restrict__ X, const float* __restrict__ W,
    const float* __restrict__ bias, float* __restrict__ Y, int Mtiles) {
  const int wavesPerBlock = blockDim.x >> 5;
  const int wave = blockIdx.x * wavesPerBlock + (threadIdx.x >> 5);
  if (wave >= Mtiles) return;  // uniform per wave -> EXEC stays all-ones
  const int lane  = threadIdx.x & 31;
  const int lrow  = lane & 15;
  const int khalf = lane >> 4;  // 0 or 1
  const int m0    = wave * 16;
  const int NOUT  = NT * 16;

  v8f acc[NT] = {};

  const float* xrow = X + (size_t)(m0 + lrow) * KD;

  for (int k0 = 0; k0 < KD; k0 += 4) {
    const int ka = k0 + 2 * khalf;
    v2f a = *(const v2f*)(xrow + ka);
    if (RELU_IN) {
      a.x = fmaxf(a.x, 0.0f);
      a.y = fmaxf(a.y, 0.0f);
    }
#pragma unroll
    for (int t = 0; t < NT; ++t) {
      const int n = t * 16 + lrow;
      const v2f b = *(const v2f*)(W + (size_t)n * KD + ka);
      acc[t] = __builtin_amdgcn_wmma_f32_16x16x4_f32(
          /*neg_a=*/false, a, /*neg_b=*/false, b,
          /*c_mod=*/(short)0, acc[t], /*reuse_a=*/false, /*reuse_b=*/false);
    }
  }

#pragma unroll
  for (int t = 0; t < NT; ++t) {
    const int col = t * 16 + lrow;
    float bv;
    if (BIAS) bv = bias[col];
#pragma unroll
    for (int r = 0; r < 8; ++r) {
      const int rowo = m0 + r + 8 * khalf;
      float v = acc[t][r];
      if (BIAS) v += bv;
      if (RELU_OUT) v = fmaxf(v, 0.0f);
      Y[(size_t)rowo * NOUT + col] = v;
    }
  }
}

// ---------------------------------------------------------------------------
// Edge scatter-add: U[dst[e]][:] += T[src[e]][:] * ew[e]
// One wave per edge; each lane owns 4 consecutive channels (float4).
// ---------------------------------------------------------------------------
__global__ __launch_bounds__(256) void scatter_add_kernel(
    const float* __restrict__ T, const int* __restrict__ src,
    const int* __restrict__ dst, const float* __restrict__ ew,
    float* __restrict__ U, int E) {
  const int wpb  = blockDim.x >> 5;
  const int wid  = blockIdx.x * wpb + (threadIdx.x >> 5);
  const int lane = threadIdx.x & 31;
  const int tot  = gridDim.x * wpb;
  for (int e = wid; e < E; e += tot) {
    const int s = src[e];
    const int d = dst[e];
    const float w = ew[e];
    const float4 v = *(const float4*)(T + (size_t)s * HH + lane * 4);
    float* up = U + (size_t)d * HH + lane * 4;
    __hip_atomic_fetch_add(up + 0, v.x * w, __ATOMIC_RELAXED, __HIP_MEMORY_SCOPE_AGENT);
    __hip_atomic_fetch_add(up + 1, v.y * w, __ATOMIC_RELAXED, __HIP_MEMORY_SCOPE_AGENT);
    __hip_atomic_fetch_add(up + 2, v.z * w, __ATOMIC_RELAXED, __HIP_MEMORY_SCOPE_AGENT);
    __hip_atomic_fetch_add(up + 3, v.w * w, __ATOMIC_RELAXED, __HIP_MEMORY_SCOPE_AGENT);
  }
}

// ---------------------------------------------------------------------------
// Final: h (N x 64) -> (B,P,64); starts = [:32], dests = [32:].
// probs[b][i] = sum_j sum_c start[b, act[i,j,0], c] * dest[b, act[i,j,1], c]
// out = softmax over i (A=50) per b. Single block.
// ---------------------------------------------------------------------------
__global__ __launch_bounds__(256) void finalize_kernel(
    const float* __restrict__ h, const int* __restrict__ actions,
    float* __restrict__ out) {
  __shared__ float probs[BB * AA];
  const int tid = threadIdx.x;
  if (tid < BB * AA) {
    const int b = tid / AA;
    const int i = tid % AA;
    float p = 0.0f;
    for (int j = 0; j < KK; ++j) {
      const int si = actions[(i * KK + j) * 2 + 0];
      const int di = actions[(i * KK + j) * 2 + 1];
      const float* sv = h + (size_t)(b * PP + si) * AV_;
      const float* dv = h + (size_t)(b * PP + di) * AV_ + HALF_;
      for (int c = 0; c < HALF_; ++c) p += sv[c] * dv[c];
    }
    probs[tid] = p;
  }
  __syncthreads();
  if (tid < BB) {
    const int b = tid;
    float mx = -1e30f;
    for (int i = 0; i < AA; ++i) mx = fmaxf(mx, probs[b * AA + i]);
    float sum = 0.0f;
    for (int i = 0; i < AA; ++i) sum += expf(probs[b * AA + i] - mx);
    const float inv = 1.0f / sum;
    for (int i = 0; i < AA; ++i)
      out[b * AA + i] = expf(probs[b * AA + i] - mx) * inv;
  }
}

extern "C" void kernel_launch(void* const* d_in, const int* in_sizes, int n_in,
                              void* d_out, int out_size, void* d_ws, size_t ws_size,
                              hipStream_t stream) {
  const int*   actions    = (const int*)d_in[0];
  const float* x          = (const float*)d_in[1];
  const int*   adj        = (const int*)d_in[2];
  const float* adj_attr   = (const float*)d_in[3];
  const int*   pass_adj   = (const int*)d_in[4];
  const float* pass_attr  = (const float*)d_in[5];
  const int*   train_adj  = (const int*)d_in[6];
  const float* train_attr = (const float*)d_in[7];
  // d_in[8] = batch (unused by the reference computation)
  const float* w1_rel  = (const float*)d_in[9];
  const float* w1_root = (const float*)d_in[10];
  const float* b1      = (const float*)d_in[11];
  const float* w2_rel  = (const float*)d_in[12];
  const float* w2_root = (const float*)d_in[13];
  const float* b2      = (const float*)d_in[14];
  const float* w3_rel  = (const float*)d_in[15];
  const float* w3_root = (const float*)d_in[16];
  const float* b3      = (const float*)d_in[17];
  const float* w_lin2  = (const float*)d_in[18];
  const float* b_lin2  = (const float*)d_in[19];
  const float* w_s0    = (const float*)d_in[20];
  const float* b_s0    = (const float*)d_in[21];
  const float* w_s1    = (const float*)d_in[22];
  const float* b_s1    = (const float*)d_in[23];
  const float* w_lin1  = (const float*)d_in[24];
  const float* b_lin1  = (const float*)d_in[25];

  float* W1 = (float*)d_ws;                       // N x H
  float* W2 = W1 + (size_t)NN * HH;               // N x H
  float* W3 = W2 + (size_t)NN * HH;               // N x H (also holds N x 64 at the end)

  const int Mtiles = NN / 16;                     // 6250
  const dim3 gblk(256);
  const dim3 ggrd((Mtiles + 7) / 8);              // 8 waves / block
  const dim3 sgrd((EE + 7) / 8);                  // 1 wave / edge

  // conv1: t=W1 = x@w1_rel^T ; u=W2 = x@w1_root^T + b1 ; scatter
  // (h1's ReLU is fused into conv2's A-loads)
  gemm_wmma_kernel<FF, HH / 16, false, false, false><<<ggrd, gblk, 0, stream>>>(x, w1_rel, nullptr, W1, Mtiles);
  gemm_wmma_kernel<FF, HH / 16, true, false, false><<<ggrd, gblk, 0, stream>>>(x, w1_root, b1, W2, Mtiles);
  scatter_add_kernel<<<sgrd, gblk, 0, stream>>>(W1, train_adj, train_adj + EE, train_attr, W2, EE);

  // conv2: in=relu(W2) ; t=W1 ; u=W3
  gemm_wmma_kernel<HH, HH / 16, false, true, false><<<ggrd, gblk, 0, stream>>>(W2, w2_rel, nullptr, W1, Mtiles);
  gemm_wmma_kernel<HH, HH / 16, true, true, false><<<ggrd, gblk, 0, stream>>>(W2, w2_root, b2, W3, Mtiles);
  scatter_add_kernel<<<sgrd, gblk, 0, stream>>>(W1, adj, adj + EE, adj_attr, W3, EE);

  // conv3: in=relu(W3) ; t=W1 ; u=W2
  gemm_wmma_kernel<HH, HH / 16, false, true, false><<<ggrd, gblk, 0, stream>>>(W3, w3_rel, nullptr, W1, Mtiles);
  gemm_wmma_kernel<HH, HH / 16, true, true, false><<<ggrd, gblk, 0, stream>>>(W3, w3_root, b3, W2, Mtiles);
  scatter_add_kernel<<<sgrd, gblk, 0, stream>>>(W1, pass_adj, pass_adj + EE, pass_attr, W2, EE);

  // lin2: h4 = relu(relu(W2) @ w_lin2^T + b_lin2)  -> W1
  gemm_wmma_kernel<HH, HH / 16, true, true, true><<<ggrd, gblk, 0, stream>>>(W2, w_lin2, b_lin2, W1, Mtiles);
  // s0: W1 -> W3
  gemm_wmma_kernel<HH, HH / 16, true, false, false><<<ggrd, gblk, 0, stream>>>(W1, w_s0, b_s0, W3, Mtiles);
  // s1: W3 -> W1
  gemm_wmma_kernel<HH, HH / 16, true, false, false><<<ggrd, gblk, 0, stream>>>(W3, w_s1, b_s1, W1, Mtiles);
  // lin1: W1 -> W3 (N x 64)
  gemm_wmma_kernel<HH, AV_ / 16, true, false, false><<<ggrd, gblk, 0, stream>>>(W1, w_lin1, b_lin1, W3, Mtiles);

  // gather + dot + softmax
  finalize_kernel<<<dim3(1), gblk, 0, stream>>>(W3, actions, (float*)d_out);
}